// RadarSecondStageGenerator_55018531062260
// MI455X (gfx1250) — compile-verified
//
#include <hip/hip_runtime.h>

typedef __attribute__((ext_vector_type(2))) float v2f;
typedef __attribute__((ext_vector_type(8))) float v8f;

#define HH 128
#define WW 128

__device__ __forceinline__ v8f wmma4(v2f a, v2f b, v8f c) {
  // D(16x16,f32) = A(16x4,f32) x B(4x16,f32) + C   -> v_wmma_f32_16x16x4_f32
  return __builtin_amdgcn_wmma_f32_16x16x4_f32(false, a, false, b, (short)0, c,
                                               false, false);
}

// ---------------------------------------------------------------------------
// Kernel 1: conv_in  (24ch -> 8ch, 3x3, pad 1), implicit GEMM on WMMA.
//   Block = one (batch, row). 8 waves, each wave -> 16 pixels x 16 oc (8 valid).
//   K ordered tap-major (kk = tap*24 + ch); the tap/ch loops are fully
//   unrolled so every LDS access is base VGPR + immediate offset.
// ---------------------------------------------------------------------------
__global__ __launch_bounds__(256) void conv_in_kernel(
    const float* __restrict__ radar, const float* __restrict__ pred,
    const float* __restrict__ w, const float* __restrict__ bias,
    float* __restrict__ xbuf) {
  __shared__ __align__(16) float s_in[3 * 130 * 24];  // [(r*130+col)*24+ch]
  __shared__ __align__(16) float s_a[216 * 16];       // [(kk>>1)*32 + m*2 + (kk&1)]

  const int tid = threadIdx.x;
  const int b = blockIdx.x >> 7, y = blockIdx.x & 127;

  // Stage input halo (rows y-1..y+1, zero-padded), channels = concat(radar, pred)
  for (int idx = tid; idx < 3 * 130 * 24; idx += 256) {
    int ch = idx % 24, rem = idx / 24;
    int col = rem % 130, r = rem / 130;
    int gy = y + r - 1, gx = col - 1;
    float v = 0.f;
    if (gy >= 0 && gy < HH && gx >= 0 && gx < WW) {
      const float* src = (ch < 12) ? (radar + (size_t)(b * 12 + ch) * (HH * WW))
                                   : (pred + (size_t)(b * 12 + ch - 12) * (HH * WW));
      v = src[gy * WW + gx];
    }
    s_in[idx] = v;
  }
  // Stage A (weights) K-pair packed, rows 8..15 zero padding.
  for (int idx = tid; idx < 216 * 16; idx += 256) {
    int p = idx >> 5, r2 = idx & 31;
    int m = r2 >> 1, s = r2 & 1;
    int kk = 2 * p + s;
    int tap = kk / 24, ch = kk - tap * 24;  // kk = tap*24 + ch
    s_a[idx] = (m < 8) ? w[m * 216 + ch * 9 + tap] : 0.f;
  }
  __syncthreads();

  const int lane = tid & 31, wave = tid >> 5;
  const int half = lane >> 4, mn = lane & 15;
  const int x0 = wave * 16;

  // Constant-offset bases: all loop offsets below are compile-time immediates.
  const float* abase = &s_a[half * 32 + mn * 2];          // + (tap*12+c4/2)*32
  const float* bbase = &s_in[(x0 + mn) * 24 + 2 * half];  // + ((dy*130+dx)*24 + c4)

  v8f acc = {0.f, 0.f, 0.f, 0.f, 0.f, 0.f, 0.f, 0.f};
#pragma unroll
  for (int tap = 0; tap < 9; ++tap) {
    const int dy = tap / 3, dx = tap - dy * 3;
#pragma unroll
    for (int c4 = 0; c4 < 24; c4 += 4) {
      v2f a = *(const v2f*)(abase + (tap * 12 + (c4 >> 1)) * 32);
      v2f bb = *(const v2f*)(bbase + (dy * 130 + dx) * 24 + c4);
      acc = wmma4(a, bb, acc);
    }
  }

  // D: lane half==0 holds oc 0..7 (valid); half==1 holds padding rows.
  if (half == 0) {
    float* outp = xbuf + (((size_t)b * 8) * HH + y) * WW + x0 + mn;
#pragma unroll
    for (int i = 0; i < 8; ++i) outp[(size_t)i * HH * WW] = acc[i] + bias[i];
  }
}

// ---------------------------------------------------------------------------
// Kernel 2: i2h conv (8->24, 3x3) + GRU gating (h0==0 folds h2h -> b_ret) +
//           conv_out (8->12, 1x1). All matrix work on WMMA f32.
// ---------------------------------------------------------------------------
__global__ __launch_bounds__(256) void gru_out_kernel(
    const float* __restrict__ xbuf, const float* __restrict__ w_i2h,
    const float* __restrict__ b_i2h, const float* __restrict__ b_ret,
    const float* __restrict__ w_out, const float* __restrict__ b_out,
    float* __restrict__ out) {
  __shared__ __align__(16) float s_x[3 * 130 * 8];   // [(r*130+col)*8+ch]
  __shared__ __align__(16) float s_a0[72 * 16];      // oc 0..15
  __shared__ __align__(16) float s_a1[72 * 16];      // oc 16..23 (+pad)
  __shared__ __align__(16) float s_a2[8 * 16];       // conv_out, oc 0..11 (+pad)
  __shared__ __align__(16) float s_i2h[8 * 24 * 16]; // [wave][oc][n]
  __shared__ __align__(16) float s_h[8 * 16 * 8];    // [wave][n][j]
  __shared__ float s_bias[24 + 24 + 12];             // b_i2h | b_ret | b_out

  const int tid = threadIdx.x;
  const int b = blockIdx.x >> 7, y = blockIdx.x & 127;

  for (int idx = tid; idx < 3 * 130 * 8; idx += 256) {
    int ch = idx & 7, rem = idx >> 3;
    int col = rem % 130, r = rem / 130;
    int gy = y + r - 1, gx = col - 1;
    float v = 0.f;
    if (gy >= 0 && gy < HH && gx >= 0 && gx < WW)
      v = xbuf[(((size_t)b * 8 + ch) * HH + gy) * WW + gx];
    s_x[idx] = v;
  }
  for (int idx = tid; idx < 72 * 16 * 2; idx += 256) {
    int t = idx / (72 * 16);
    int id2 = idx - t * 72 * 16;
    int p = id2 >> 5, r2 = id2 & 31;
    int m = r2 >> 1, s = r2 & 1;
    int kk = 2 * p + s;
    int tap = kk >> 3, ch = kk & 7;  // kk = tap*8 + ch
    int oc = t * 16 + m;
    float v = (oc < 24) ? w_i2h[oc * 72 + ch * 9 + tap] : 0.f;
    (t ? s_a1 : s_a0)[id2] = v;
  }
  if (tid < 8 * 16) {
    int p = tid >> 5, r2 = tid & 31;
    int m = r2 >> 1, s = r2 & 1;
    int kk = 2 * p + s;
    s_a2[tid] = (m < 12) ? w_out[m * 8 + kk] : 0.f;
  }
  if (tid < 60) {
    float v;
    if (tid < 24) v = b_i2h[tid];
    else if (tid < 48) v = b_ret[tid - 24];
    else v = b_out[tid - 48];
    s_bias[tid] = v;
  }
  __syncthreads();

  const int lane = tid & 31, wave = tid >> 5;
  const int half = lane >> 4, mn = lane & 15;
  const int x0 = wave * 16;

  // --- i2h: K = 72 (tap-major x 8 ch), two M tiles (oc 0..15, 16..31) ---
  const float* a0base = &s_a0[half * 32 + mn * 2];
  const float* a1base = &s_a1[half * 32 + mn * 2];
  const float* bbase = &s_x[(x0 + mn) * 8 + 2 * half];

  v8f c0 = {0.f, 0.f, 0.f, 0.f, 0.f, 0.f, 0.f, 0.f};
  v8f c1 = {0.f, 0.f, 0.f, 0.f, 0.f, 0.f, 0.f, 0.f};
#pragma unroll
  for (int tap = 0; tap < 9; ++tap) {
    const int dy = tap / 3, dx = tap - dy * 3;
#pragma unroll
    for (int c4 = 0; c4 < 8; c4 += 4) {
      v2f bb = *(const v2f*)(bbase + (dy * 130 + dx) * 8 + c4);
      v2f a0 = *(const v2f*)(a0base + (tap * 4 + (c4 >> 1)) * 32);
      v2f a1 = *(const v2f*)(a1base + (tap * 4 + (c4 >> 1)) * 32);
      c0 = wmma4(a0, bb, c0);
      c1 = wmma4(a1, bb, c1);
    }
  }

  // Spill i2h tile to per-wave LDS so gating can read across lane halves.
  float* scr = &s_i2h[wave * 24 * 16];
#pragma unroll
  for (int i = 0; i < 8; ++i) scr[(i + 8 * half) * 16 + mn] = c0[i];  // oc 0..15
  if (half == 0) {
#pragma unroll
    for (int i = 0; i < 8; ++i) scr[(16 + i) * 16 + mn] = c1[i];      // oc 16..23
  }
  __syncthreads();

  // --- gating: h = (1-u) * leaky(i_m + r*rb_m); h0==0 so h2h == b_ret ---
  float* hrow = &s_h[wave * 16 * 8];  // [n][j], K-contiguous for conv_out B
#pragma unroll
  for (int q = 0; q < 4; ++q) {
    int e = lane + 32 * q;            // 128 (ch,pixel) elems per wave
    int j = e >> 4, p = e & 15;
    float ir = scr[j * 16 + p] + s_bias[j] + s_bias[24 + j];
    float iu = scr[(8 + j) * 16 + p] + s_bias[8 + j] + s_bias[24 + 8 + j];
    float im = scr[(16 + j) * 16 + p] + s_bias[16 + j];
    float r = 1.f / (1.f + __expf(-ir));
    float u = 1.f / (1.f + __expf(-iu));
    float mpre = im + r * s_bias[24 + 16 + j];
    float mem = (mpre >= 0.f) ? mpre : 0.2f * mpre;
    hrow[p * 8 + j] = (1.f - u) * mem;
  }
  __syncthreads();

  // --- conv_out 1x1: D(12x16) = W(12x8) x h(8x16), K=8 -> 2 WMMAs ---
  v8f c2 = {0.f, 0.f, 0.f, 0.f, 0.f, 0.f, 0.f, 0.f};
#pragma unroll
  for (int k4 = 0; k4 < 8; k4 += 4) {
    int kk = k4 + 2 * half;
    v2f a2 = *(const v2f*)&s_a2[(kk >> 1) * 32 + mn * 2];
    v2f bb = *(const v2f*)&hrow[mn * 8 + kk];
    c2 = wmma4(a2, bb, c2);
  }

  size_t base = (((size_t)b * 12) * HH + y) * WW + x0 + mn;
  if (half == 0) {
#pragma unroll
    for (int i = 0; i < 8; ++i)
      out[base + (size_t)i * (HH * WW)] = c2[i] + s_bias[48 + i];
  } else {
#pragma unroll
    for (int i = 0; i < 4; ++i)
      out[base + (size_t)(8 + i) * (HH * WW)] = c2[i] + s_bias[48 + 8 + i];
  }
}

// ---------------------------------------------------------------------------
extern "C" void kernel_launch(void* const* d_in, const int* in_sizes, int n_in,
                              void* d_out, int out_size, void* d_ws,
                              size_t ws_size, hipStream_t stream) {
  const float* radar = (const float*)d_in[0];
  const float* pred = (const float*)d_in[1];
  const float* w_ci = (const float*)d_in[2];
  const float* b_ci = (const float*)d_in[3];
  const float* w_i2h = (const float*)d_in[4];
  const float* b_i2h = (const float*)d_in[5];
  // d_in[6..12] (i2f/h2f/flows weights, w_ret) are dead: h0 == 0 makes the
  // warped hidden state identically zero, so h2h reduces to b_ret.
  const float* b_ret = (const float*)d_in[13];
  const float* w_out = (const float*)d_in[14];
  const float* b_out = (const float*)d_in[15];

  float* xbuf = (float*)d_ws;  // 32*8*128*128 f32 = 16 MB (fits ws, lives in L2)

  dim3 grid(32 * 128), block(256);
  conv_in_kernel<<<grid, block, 0, stream>>>(radar, pred, w_ci, b_ci, xbuf);
  gru_out_kernel<<<grid, block, 0, stream>>>(xbuf, w_i2h, b_i2h, b_ret, w_out,
                                             b_out, (float*)d_out);
}